// Attention_352187318644
// MI455X (gfx1250) — compile-verified
//
#include <hip/hip_runtime.h>
#include <hip/hip_bf16.h>

// Problem constants (from reference)
#define BB     128
#define HH     32
#define DD     128
#define SS     64
#define START  32
#define SCALE  0.125f

typedef __attribute__((ext_vector_type(2))) float v2f;
typedef __attribute__((ext_vector_type(8))) float v8f;

// ---------------------------------------------------------------------------
// Kernel P: per head h, per batch-group of 16:
//   Q = X @ Wq, Knew = X @ Wk, Vnew = X @ Wv   (all [16,128] = [16,128]x[128,128])
// using fp32 WMMA 16x16x4. Knew/Vnew scatter into the live k/v cache at gen_id.
// ---------------------------------------------------------------------------
__global__ __launch_bounds__(256) void proj_qkv(
    const float* __restrict__ x,    // [B,H,D]
    const float* __restrict__ wq,   // [H,D,D]
    const float* __restrict__ wk,
    const float* __restrict__ wv,
    float* __restrict__ qbuf,       // [B,H,D]
    float* __restrict__ kcache,     // [B,H,S,D]
    float* __restrict__ vcache,     // [B,H,S,D]
    int gen_id)
{
    const int h  = blockIdx.x;           // head
    const int b0 = blockIdx.y * 16;      // batch tile start

    __shared__ float Xs[16 * DD];        // X tile, row major [m][d]

    for (int i = threadIdx.x; i < 16 * DD; i += 256) {
        int m = i >> 7, d = i & 127;
        Xs[i] = x[((size_t)(b0 + m) * HH + h) * DD + d];
    }
    __syncthreads();

    const int wave = threadIdx.x >> 5;   // 0..7 -> output column tile
    const int lane = threadIdx.x & 31;
    const int n0   = wave * 16;
    const int nl   = lane & 15;          // A-row index (M) and C/D column (N)
    const int kh   = lane >> 4;          // half-wave: selects K pair

    const float* Ws[3] = { wq + (size_t)h * DD * DD,
                           wk + (size_t)h * DD * DD,
                           wv + (size_t)h * DD * DD };

    #pragma unroll
    for (int j = 0; j < 3; ++j) {
        const float* W = Ws[j];
        v8f c = {};
        #pragma unroll 4
        for (int k0 = 0; k0 < DD; k0 += 4) {
            v2f a, b;
            // A 16x4 f32: lanes 0-15 hold K={0,1}, lanes 16-31 hold K={2,3}
            a.x = Xs[nl * DD + k0 + 2 * kh + 0];
            a.y = Xs[nl * DD + k0 + 2 * kh + 1];
            // B 4x16 f32: vgpr0 -> K = k0+kh, vgpr1 -> K = k0+2+kh, N across lanes
            b.x = W[(size_t)(k0 + kh)     * DD + n0 + nl];
            b.y = W[(size_t)(k0 + 2 + kh) * DD + n0 + nl];
            c = __builtin_amdgcn_wmma_f32_16x16x4_f32(
                    false, a, false, b, (short)0, c, false, false);
        }
        // C/D layout: VGPR r -> M = r (lanes 0-15) or r+8 (lanes 16-31); N = lane&15
        #pragma unroll
        for (int r = 0; r < 8; ++r) {
            int mm = r + 8 * kh;
            float val = c[r];
            size_t bh = (size_t)(b0 + mm) * HH + h;
            if (j == 0)      qbuf[bh * DD + n0 + nl] = val;
            else if (j == 1) kcache[(bh * SS + gen_id) * DD + n0 + nl] = val;
            else             vcache[(bh * SS + gen_id) * DD + n0 + nl] = val;
        }
    }
}

// ---------------------------------------------------------------------------
// Kernel A: per head h, per batch-group of 16:
//   scores = K . q * SCALE -> softmax over full S=64 -> ctx = p . V
//   x_new  = ctx @ Wo  (fp32 WMMA)
// ---------------------------------------------------------------------------
__global__ __launch_bounds__(256) void attn_out(
    const float* __restrict__ qbuf,     // [B,H,D]
    const float* __restrict__ kcache,   // [B,H,S,D]
    const float* __restrict__ vcache,   // [B,H,S,D]
    const float* __restrict__ wo,       // [H,D,D]
    float* __restrict__ xbuf)           // [B,H,D] (written in place)
{
    const int h  = blockIdx.x;
    const int b0 = blockIdx.y * 16;

    __shared__ float Qs[16 * DD];       // q tiles
    __shared__ float Ps[16 * SS];       // scores -> probs
    __shared__ float Cs[16 * DD];       // context

    for (int i = threadIdx.x; i < 16 * DD; i += 256) {
        int m = i >> 7, d = i & 127;
        Qs[i] = qbuf[((size_t)(b0 + m) * HH + h) * DD + d];
    }
    __syncthreads();

    const int bl = threadIdx.x >> 4;    // local batch 0..15
    const int g  = threadIdx.x & 15;    // sub-worker 0..15

    // ---- Phase 1: raw scores (4 cache rows per thread, full 128-d dot) ----
    {
        const float* kb = kcache + ((size_t)(b0 + bl) * HH + h) * SS * DD;
        const float4* qr = (const float4*)(Qs + bl * DD);
        #pragma unroll
        for (int i = 0; i < 4; ++i) {
            int s = g * 4 + i;
            const float4* kr = (const float4*)(kb + (size_t)s * DD);
            float acc = 0.f;
            #pragma unroll
            for (int d4 = 0; d4 < DD / 4; ++d4) {
                float4 kv = kr[d4], qv = qr[d4];
                acc += kv.x * qv.x + kv.y * qv.y + kv.z * qv.z + kv.w * qv.w;
            }
            Ps[bl * SS + s] = acc * SCALE;
        }
    }
    __syncthreads();

    // ---- Softmax over S=64 (redundant per-thread reduce, then normalize) ----
    float mx = -3.402823466e+38f;
    for (int s = 0; s < SS; ++s) mx = fmaxf(mx, Ps[bl * SS + s]);
    float sum = 0.f;
    for (int s = 0; s < SS; ++s) sum += __expf(Ps[bl * SS + s] - mx);
    float inv = 1.f / sum;
    float p4[4];
    #pragma unroll
    for (int i = 0; i < 4; ++i)
        p4[i] = __expf(Ps[bl * SS + g * 4 + i] - mx) * inv;
    __syncthreads();                    // all raw-score reads done
    #pragma unroll
    for (int i = 0; i < 4; ++i) Ps[bl * SS + g * 4 + i] = p4[i];
    __syncthreads();                    // probs visible

    // ---- Phase 2: ctx[bl, d] = sum_s p[s] * v[s, d]  (8 d's per thread) ----
    {
        const float* vb = vcache + ((size_t)(b0 + bl) * HH + h) * SS * DD;
        const int c8 = g * 8;
        float acc[8] = {0.f, 0.f, 0.f, 0.f, 0.f, 0.f, 0.f, 0.f};
        for (int s = 0; s < SS; ++s) {
            float p = Ps[bl * SS + s];
            const float4* vr = (const float4*)(vb + (size_t)s * DD + c8);
            float4 a = vr[0], bq = vr[1];
            acc[0] += p * a.x;  acc[1] += p * a.y;
            acc[2] += p * a.z;  acc[3] += p * a.w;
            acc[4] += p * bq.x; acc[5] += p * bq.y;
            acc[6] += p * bq.z; acc[7] += p * bq.w;
        }
        #pragma unroll
        for (int i = 0; i < 8; ++i) Cs[bl * DD + c8 + i] = acc[i];
    }
    __syncthreads();

    // ---- Phase 3: x_new = ctx @ Wo via fp32 WMMA ----
    {
        const int wave = threadIdx.x >> 5;
        const int lane = threadIdx.x & 31;
        const int n0   = wave * 16;
        const int nl   = lane & 15;
        const int kh   = lane >> 4;
        const float* W = wo + (size_t)h * DD * DD;

        v8f c = {};
        #pragma unroll 4
        for (int k0 = 0; k0 < DD; k0 += 4) {
            v2f a, b;
            a.x = Cs[nl * DD + k0 + 2 * kh + 0];
            a.y = Cs[nl * DD + k0 + 2 * kh + 1];
            b.x = W[(size_t)(k0 + kh)     * DD + n0 + nl];
            b.y = W[(size_t)(k0 + 2 + kh) * DD + n0 + nl];
            c = __builtin_amdgcn_wmma_f32_16x16x4_f32(
                    false, a, false, b, (short)0, c, false, false);
        }
        #pragma unroll
        for (int r = 0; r < 8; ++r) {
            int mm = r + 8 * kh;
            xbuf[((size_t)(b0 + mm) * HH + h) * DD + n0 + nl] = c[r];
        }
    }
}

// ---------------------------------------------------------------------------
// Host-side orchestration: seed k/v caches (in d_out) and x (in workspace)
// via async d2d copies, then 32 steps x {proj, attn}. Final x copied to d_out.
// ---------------------------------------------------------------------------
extern "C" void kernel_launch(void* const* d_in, const int* in_sizes, int n_in,
                              void* d_out, int out_size, void* d_ws, size_t ws_size,
                              hipStream_t stream) {
    (void)in_sizes; (void)n_in; (void)out_size; (void)ws_size;

    const float* x_in = (const float*)d_in[0];
    const float* k_in = (const float*)d_in[1];
    const float* v_in = (const float*)d_in[2];
    const float* wq   = (const float*)d_in[3];
    const float* wk   = (const float*)d_in[4];
    const float* wv   = (const float*)d_in[5];
    const float* wo   = (const float*)d_in[6];

    const size_t KV  = (size_t)BB * HH * SS * DD;   // k (or v) cache elements
    const size_t BHD = (size_t)BB * HH * DD;        // x / q elements

    float* k_out = (float*)d_out;                   // output k cache (live)
    float* v_out = k_out + KV;                      // output v cache (live)
    float* x_out = v_out + KV;                      // final x

    float* xbuf = (float*)d_ws;                     // [B,H,D] working x
    float* qbuf = xbuf + BHD;                       // [B,H,D] working q

    // Seed live caches and working x (d2d, graph-capture safe)
    hipMemcpyAsync(k_out, k_in, KV  * sizeof(float), hipMemcpyDeviceToDevice, stream);
    hipMemcpyAsync(v_out, v_in, KV  * sizeof(float), hipMemcpyDeviceToDevice, stream);
    hipMemcpyAsync(xbuf,  x_in, BHD * sizeof(float), hipMemcpyDeviceToDevice, stream);

    dim3 grid(HH, BB / 16);   // (32, 8)
    for (int gen = START; gen < SS; ++gen) {
        proj_qkv<<<grid, 256, 0, stream>>>(xbuf, wq, wk, wv, qbuf, k_out, v_out, gen);
        attn_out<<<grid, 256, 0, stream>>>(qbuf, k_out, v_out, wo, xbuf);
    }

    hipMemcpyAsync(x_out, xbuf, BHD * sizeof(float), hipMemcpyDeviceToDevice, stream);
}